// YoloLoss_11836929868144
// MI455X (gfx1250) — compile-verified
//
#include <hip/hip_runtime.h>

// ---------------------------------------------------------------------------
// YOLOv3 loss for MI455X (gfx1250, wave32).
// Memory-bound streaming kernel: ~232 MB single-use traffic -> ~10us roofline
// at 23.3 TB/s. One thread per (b, anchor, y, x) cell; y_pred reads are
// coalesced across lanes (channel stride = h*w), y_true is a contiguous 340B
// per-thread stream; both use non-temporal loads (single-use, > L2 capacity).
// Reduction: per-wave V_WMMA_F32_16X16X4_F32 (A = ones => column sums fold
// lane n with lane n+16) + 4 shfl_xor steps, block partials to workspace,
// deterministic single-block finalize (no float atomics -> replay-stable).
// ---------------------------------------------------------------------------

typedef __attribute__((ext_vector_type(2))) float v2f;
typedef __attribute__((ext_vector_type(8))) float v8f;

#define EPSF 1e-7f

__device__ __forceinline__ float nt_load(const float* p) {
  return __builtin_nontemporal_load(p);
}

__device__ __forceinline__ float sigmoidf_(float x) {
  return 1.0f / (1.0f + __expf(-x));
}

// BCE on an (already sigmoided) probability, with reference's eps clipping.
__device__ __forceinline__ float bcef_(float p, float t) {
  p = fminf(fmaxf(p, EPSF), 1.0f - EPSF);
  return -t * __logf(p) - (1.0f - t) * __logf(1.0f - p);
}

// Wave32 sum reduction seeded by one V_WMMA_F32_16X16X4_F32.
// A = all-ones (both A VGPRs = 1.0 covers every (m,k)), B VGPR0 = x per lane,
// B VGPR1 = 0. Then D[m][n] = sum_k B[k][n] = x_n + x_{n+16} regardless of
// which two K-rows the lane-halves of B VGPR0 map to. Every lane of D[0]
// holds y_{lane&15}; 4 xor-shuffles finish the 16-way sum (all lanes end with
// the wave total). EXEC must be all 1s at the call site.
__device__ __forceinline__ float waveReduce32(float x) {
  v2f A; A[0] = 1.0f; A[1] = 1.0f;
  v2f B; B[0] = x;    B[1] = 0.0f;
  v8f C = {0.f, 0.f, 0.f, 0.f, 0.f, 0.f, 0.f, 0.f};
  v8f D = __builtin_amdgcn_wmma_f32_16x16x4_f32(
      /*neg_a=*/false, A, /*neg_b=*/false, B,
      /*c_mod=*/(short)0, C, /*reuse_a=*/false, /*reuse_b=*/false);
  float v = D[0];
  v += __shfl_xor(v, 1, 32);
  v += __shfl_xor(v, 2, 32);
  v += __shfl_xor(v, 4, 32);
  v += __shfl_xor(v, 8, 32);
  return v;
}

// 256-thread block reduction; thread 0 writes the block sum to *dst.
__device__ __forceinline__ void blockReduceStore(float local, float* dst) {
  __shared__ float sbuf[8];
  float w = waveReduce32(local);
  const int lane = threadIdx.x & 31;
  const int wid  = threadIdx.x >> 5;
  if (lane == 0) sbuf[wid] = w;
  __syncthreads();
  float r = (threadIdx.x < 8) ? sbuf[threadIdx.x] : 0.0f;
  r += __shfl_xor(r, 1, 32);
  r += __shfl_xor(r, 2, 32);
  r += __shfl_xor(r, 4, 32);
  if (threadIdx.x == 0) *dst = r;
}

// One detection head. yp: (32, 255, g, g) f32; yt: (32, 3, g, g, 85) f32;
// tgt: (32, 20, 5) f32. Writes one partial sum per block.
__global__ __launch_bounds__(256) void yolo_layer_kernel(
    const float* __restrict__ yp, const float* __restrict__ yt,
    const float* __restrict__ tgt, float* __restrict__ partials, int gsz,
    float a0x, float a0y, float a1x, float a1y, float a2x, float a2y) {
  const int hw = gsz * gsz;
  const int N  = 32 * 3 * hw;
  const float inv_g = 1.0f / (float)gsz;

  const int i = blockIdx.x * 256 + threadIdx.x;
  float local = 0.0f;
  if (i < N) {
    const int xy = i % hw;
    const int ba = i / hw;          // b*3 + a
    const int a  = ba % 3;
    const int b  = ba / 3;
    const int gx = xy % gsz;
    const int gy = xy / gsz;

    const float* prow = yp + (size_t)(b * 255 + a * 85) * (size_t)hw + xy;
    const float* trow = yt + (size_t)i * 85;

    const float r0 = nt_load(prow);                    // tx raw
    const float r1 = nt_load(prow + (size_t)hw);       // ty raw
    const float r2 = nt_load(prow + (size_t)2 * hw);   // tw raw
    const float r3 = nt_load(prow + (size_t)3 * hw);   // th raw
    const float r4 = nt_load(prow + (size_t)4 * hw);   // conf raw
    const float t0 = nt_load(trow + 0);
    const float t1 = nt_load(trow + 1);
    const float t2 = nt_load(trow + 2);
    const float t3 = nt_load(trow + 3);
    const float mask = nt_load(trow + 4);

    const float aw = (a == 0) ? a0x : ((a == 1) ? a1x : a2x);
    const float ah = (a == 0) ? a0y : ((a == 1) ? a1y : a2y);

    const float sx = sigmoidf_(r0);
    const float sy = sigmoidf_(r1);
    const float sconf = sigmoidf_(r4);

    // Decoded (normalized) prediction box.
    const float bx = (sx + (float)gx) * inv_g;
    const float by = (sy + (float)gy) * inv_g;
    const float bw = __expf(r2) * aw * (1.0f / 416.0f);
    const float bh = __expf(r3) * ah * (1.0f / 416.0f);

    // Ground-truth transforms.
    const float tx = t0 * (float)gsz - (float)gx;
    const float ty = t1 * (float)gsz - (float)gy;
    float tw = __logf(t2 * 416.0f / aw);
    float th = __logf(t3 * 416.0f / ah);
    if (!(mask > 0.0f)) { tw = 0.0f; th = 0.0f; }
    const float loss_scale = 2.0f - t2 * t3;

    // Best IoU vs 20 targets of this batch (targets are L1/L2 resident).
    const float p_minx = bx - 0.5f * bw, p_maxx = bx + 0.5f * bw;
    const float p_miny = by - 0.5f * bh, p_maxy = by + 0.5f * bh;
    const float p_area = bw * bh;
    const float* tb = tgt + (size_t)b * 20 * 5;
    float best = -1e30f;
#pragma unroll 4
    for (int n = 0; n < 20; ++n) {
      const float cx = tb[n * 5 + 0], cy = tb[n * 5 + 1];
      const float ww = tb[n * 5 + 2], hh = tb[n * 5 + 3];
      float iw = fminf(p_maxx, cx + 0.5f * ww) - fmaxf(p_minx, cx - 0.5f * ww);
      float ih = fminf(p_maxy, cy + 0.5f * hh) - fmaxf(p_miny, cy - 0.5f * hh);
      iw = fmaxf(iw, 0.0f);
      ih = fmaxf(ih, 0.0f);
      const float inter = iw * ih;
      const float iou = inter / (p_area + ww * hh - inter);
      best = fmaxf(best, iou);
    }
    const float neg_mask = (best < 0.5f) ? (1.0f - mask) : 0.0f;

    const float lxy = bcef_(sx, tx) + bcef_(sy, ty);
    const float dw = r2 - tw, dh = r3 - th;
    const float lwh = 0.5f * (dw * dw + dh * dh);
    const float lconf = bcef_(sconf, mask);

    float lcls = 0.0f;
#pragma unroll 4
    for (int c = 0; c < 80; ++c) {
      const float rc = nt_load(prow + (size_t)(5 + c) * hw);
      const float tc = nt_load(trow + 5 + c);
      lcls += bcef_(sigmoidf_(rc), tc);
    }

    local = mask * loss_scale * (lxy + lwh) + (mask + neg_mask) * lconf +
            mask * lcls;
  }
  // All 256 threads reconverge here (tail threads contribute 0) so EXEC is
  // all-1s for the WMMA-based reduction.
  blockReduceStore(local, partials + blockIdx.x);
}

// Deterministic final reduction of all block partials -> scalar loss.
__global__ __launch_bounds__(256) void yolo_finalize_kernel(
    const float* __restrict__ partials, int n, float* __restrict__ out) {
  float s = 0.0f;
  for (int i = threadIdx.x; i < n; i += 256) s += partials[i];
  blockReduceStore(s, out);
}

extern "C" void kernel_launch(void* const* d_in, const int* in_sizes, int n_in,
                              void* d_out, int out_size, void* d_ws,
                              size_t ws_size, hipStream_t stream) {
  (void)in_sizes; (void)n_in; (void)out_size; (void)ws_size;
  const float* yp0 = (const float*)d_in[0];  // (32,255,13,13)
  const float* yp1 = (const float*)d_in[1];  // (32,255,26,26)
  const float* yp2 = (const float*)d_in[2];  // (32,255,52,52)
  const float* yt0 = (const float*)d_in[3];  // (32,3,13,13,85)
  const float* yt1 = (const float*)d_in[4];  // (32,3,26,26,85)
  const float* yt2 = (const float*)d_in[5];  // (32,3,52,52,85)
  const float* tgt = (const float*)d_in[6];  // (32,20,5)
  float* partials = (float*)d_ws;

  const int nb0 = (32 * 3 * 13 * 13 + 255) / 256;  // 64
  const int nb1 = (32 * 3 * 26 * 26 + 255) / 256;  // 254
  const int nb2 = (32 * 3 * 52 * 52 + 255) / 256;  // 1014

  // ANCHOR_MASK[0]=[6,7,8] -> 13x13; [3,4,5] -> 26x26; [0,1,2] -> 52x52.
  yolo_layer_kernel<<<nb0, 256, 0, stream>>>(
      yp0, yt0, tgt, partials, 13,
      116.f, 90.f, 156.f, 198.f, 373.f, 326.f);
  yolo_layer_kernel<<<nb1, 256, 0, stream>>>(
      yp1, yt1, tgt, partials + nb0, 26,
      30.f, 61.f, 62.f, 45.f, 59.f, 119.f);
  yolo_layer_kernel<<<nb2, 256, 0, stream>>>(
      yp2, yt2, tgt, partials + nb0 + nb1, 52,
      10.f, 13.f, 16.f, 30.f, 33.f, 23.f);

  yolo_finalize_kernel<<<1, 256, 0, stream>>>(partials, nb0 + nb1 + nb2,
                                              (float*)d_out);
}